// Conv2dLocal_46815143526701
// MI455X (gfx1250) — compile-verified
//
#include <hip/hip_runtime.h>
#include <stdint.h>

// Problem constants (from reference): N=8, H=512, W=512, C=3, K=5, 'same' pad=2.
#define B_N 8
#define B_H 512
#define B_W 512
#define B_C 3
#define KK 5
#define KTAPS 25
#define PADX 2

// Tile: 32x8 output pixels per 256-thread block (8 wave32 waves).
#define TILE_W 32
#define TILE_H 8
#define IN_TW (TILE_W + 2 * PADX)     // 36 input cols
#define IN_TH (TILE_H + 2 * PADX)     // 12 input rows
#define IN_ROWF (IN_TW * B_C)         // 108 floats per staged input row
#define IN_FLOATS (IN_TH * IN_ROWF)   // 1296 floats (5184 B)
#define W_ROWF (TILE_W * KTAPS)       // 800 floats per staged weight row
#define W_FLOATS (TILE_H * W_ROWF)    // 6400 floats (25600 B)
#define W_VEC4 (W_FLOATS / 4)         // 1600 float4 transfers

#ifndef __has_builtin
#define __has_builtin(x) 0
#endif

#if defined(__AMDGCN__) && __has_builtin(__builtin_amdgcn_global_load_async_to_lds_b32)
#define HAVE_ASYNC_B32 1
#else
#define HAVE_ASYNC_B32 0
#endif
#if defined(__AMDGCN__) && __has_builtin(__builtin_amdgcn_global_load_async_to_lds_b128)
#define HAVE_ASYNC_B128 1
#else
#define HAVE_ASYNC_B128 0
#endif
#if defined(__AMDGCN__) && __has_builtin(__builtin_amdgcn_s_wait_asynccnt)
#define HAVE_WAIT_ASYNC 1
#else
#define HAVE_WAIT_ASYNC 0
#endif

// gfx12-family CPol encoding: TH in bits [2:0] (1 = non-temporal), scope [4:3].
#define CPOL_RT 0
#define CPOL_NT 1

// Exact builtin parameter types (from hipcc diagnostics):
//   b32 : (AS1 int*,  AS3 int*,  imm offset, imm cpol)
//   b128: (AS1 v4i*,  AS3 v4i*,  imm offset, imm cpol)
typedef int v4i __attribute__((ext_vector_type(4)));
typedef __attribute__((address_space(1))) int  gi32;
typedef __attribute__((address_space(3))) int  li32;
typedef __attribute__((address_space(1))) v4i  gv4i;
typedef __attribute__((address_space(3))) v4i  lv4i;
typedef __attribute__((address_space(3))) void las_void;

// ---- async global -> LDS, 4 bytes per lane (regular-temporal) ------------
__device__ __forceinline__ void async_load_b32(const float* g, float* l) {
#if HAVE_ASYNC_B32
  __builtin_amdgcn_global_load_async_to_lds_b32(
      (gi32*)g, (li32*)l, /*offset=*/0, /*cpol=*/CPOL_RT);
#else
  unsigned loff = (unsigned)(uintptr_t)(las_void*)l;  // AS3 byte offset
  unsigned long long ga = (unsigned long long)(uintptr_t)g;
  asm volatile("global_load_async_to_lds_b32 %0, %1, off"
               :: "v"(loff), "v"(ga) : "memory");
#endif
}

// ---- async global -> LDS, 16 bytes per lane, non-temporal (streaming) ----
__device__ __forceinline__ void async_load_b128_nt(const float* g, float* l) {
#if HAVE_ASYNC_B128
  __builtin_amdgcn_global_load_async_to_lds_b128(
      (gv4i*)g, (lv4i*)l, /*offset=*/0, /*cpol=*/CPOL_NT);
#else
  unsigned loff = (unsigned)(uintptr_t)(las_void*)l;
  unsigned long long ga = (unsigned long long)(uintptr_t)g;
  asm volatile("global_load_async_to_lds_b128 %0, %1, off th:TH_LOAD_NT"
               :: "v"(loff), "v"(ga) : "memory");
#endif
}

__device__ __forceinline__ void wait_async_zero() {
#if HAVE_WAIT_ASYNC
  __builtin_amdgcn_s_wait_asynccnt(0);
#else
  asm volatile("s_wait_asynccnt 0" ::: "memory");
#endif
}

__global__ __launch_bounds__(256) void
conv2d_local_kernel(const float* __restrict__ in,   // (N,H,W,C)
                    const float* __restrict__ wgt,  // (N,H,W,25)
                    float* __restrict__ out) {      // (N,H,W,C)
  __shared__ __attribute__((aligned(16))) float s_w[W_FLOATS];
  __shared__ __attribute__((aligned(16))) float s_in[IN_FLOATS];

  const int tid = threadIdx.x;
  const int n   = blockIdx.z;
  const int th0 = blockIdx.y * TILE_H;
  const int tw0 = blockIdx.x * TILE_W;

  // ---------------- stage weights: 1600 x float4, fully coalesced, NT ----
  // Row start = ((n*H + th0 + r)*W + tw0)*25 floats; tw0 % 32 == 0 so the
  // byte offset is a multiple of 3200 -> 16B aligned. Weights are read-once
  // streaming traffic (210 MB total): non-temporal keeps them out of L2.
  const float* wbase = wgt + ((size_t)((n * B_H + th0) * B_W) + tw0) * KTAPS;
#pragma unroll
  for (int i = 0; i < (W_VEC4 + 255) / 256; ++i) {
    int e = i * 256 + tid;
    if (e < W_VEC4) {
      int row = e / (W_ROWF / 4);  // 0..7
      int col = e % (W_ROWF / 4);  // float4 within row
      const float* g = wbase + (size_t)row * (B_W * KTAPS) + col * 4;
      float* l = s_w + row * W_ROWF + col * 4;
      async_load_b128_nt(g, l);
    }
  }

  // ---------------- stage input halo tile: 1296 floats (RT: halo reuse) ---
  const int gh0 = th0 - PADX;
  const int gw0c = (tw0 - PADX) * B_C;  // column offset in floats within a row
#pragma unroll
  for (int i = 0; i < (IN_FLOATS + 255) / 256; ++i) {
    int e = i * 256 + tid;
    if (e < IN_FLOATS) {
      int lr  = e / IN_ROWF;  // staged row 0..11
      int rem = e % IN_ROWF;  // (col*3 + c) within row
      int gh  = gh0 + lr;
      int gc  = gw0c + rem;   // float index within the global row
      float* l = s_in + e;
      if ((unsigned)gh < (unsigned)B_H && (unsigned)gc < (unsigned)(B_W * B_C)) {
        const float* g = in + (size_t)(n * B_H + gh) * (B_W * B_C) + gc;
        async_load_b32(g, l);
      } else {
        *l = 0.0f;  // zero padding via DS store (disjoint from async writes)
      }
    }
  }

  wait_async_zero();   // wave's own async transfers into LDS complete
  __syncthreads();     // all waves' transfers visible (barrier + dscnt)

  // ---------------- compute: one output pixel (3 channels) per thread ----
  const int tx = tid & (TILE_W - 1);
  const int ty = tid >> 5;

  float wv[KTAPS];
#pragma unroll
  for (int k = 0; k < KTAPS; ++k)
    wv[k] = s_w[tid * KTAPS + k];  // stride 25 (odd) -> bank-conflict-free

  float a0 = 0.f, a1 = 0.f, a2 = 0.f;
#pragma unroll
  for (int kh = 0; kh < KK; ++kh) {
#pragma unroll
    for (int kw = 0; kw < KK; ++kw) {
      const float wk = wv[kh * KK + kw];
      const float* p = s_in + ((ty + kh) * IN_TW + (tx + kw)) * B_C;
      a0 = fmaf(p[0], wk, a0);
      a1 = fmaf(p[1], wk, a1);
      a2 = fmaf(p[2], wk, a2);
    }
  }

  // Write-once streaming output: non-temporal stores avoid L2 pollution.
  const int oh = th0 + ty, ow = tw0 + tx;
  float* o = out + ((size_t)(n * B_H + oh) * B_W + ow) * B_C;
  __builtin_nontemporal_store(a0, o + 0);
  __builtin_nontemporal_store(a1, o + 1);
  __builtin_nontemporal_store(a2, o + 2);
}

extern "C" void kernel_launch(void* const* d_in, const int* in_sizes, int n_in,
                              void* d_out, int out_size, void* d_ws, size_t ws_size,
                              hipStream_t stream) {
  const float* in  = (const float*)d_in[0];   // (8,512,512,3) fp32
  const float* wgt = (const float*)d_in[1];   // (8,512,512,25) fp32
  float* out = (float*)d_out;                 // (8,512,512,3) fp32

  dim3 grid(B_W / TILE_W, B_H / TILE_H, B_N);  // 16 x 64 x 8 = 8192 blocks
  conv2d_local_kernel<<<grid, 256, 0, stream>>>(in, wgt, out);
}